// SharedGINEncoder_68470368632924
// MI455X (gfx1250) — compile-verified
//
#include <hip/hip_runtime.h>
#include <hip/hip_bf16.h>

// ---------------------------------------------------------------------------
// SharedGINEncoder for MI455X (gfx1250).
// Scatter-bound workload (~820 MB of edge traffic vs 2.5 GFLOP of GEMM).
// GEMMs use V_WMMA_F32_16X16X4_F32 (exact fp32 semantics, wave32 layout),
// scatter uses hardware global_atomic_add_f32 with wave-per-edge coalescing.
// ---------------------------------------------------------------------------

#define N_NODES  50000
#define N_EDGES  800000
#define IN_DIM   128
#define HID      64
#define N_GRAPHS 256

typedef __attribute__((ext_vector_type(2))) float v2f;
typedef __attribute__((ext_vector_type(8))) float v8f;

// ---------------------------------------------------------------------------
// GEMM: out[n_rows x 64] = act( (A (+ AGG)) [n_rows x K] @ W [K x 64] + bias )
// Block = 256 threads = 8 waves; each wave computes one 16x16 tile of C.
// Block tile: 32 rows x 64 cols.  W + bias staged in LDS, shared by all waves.
// fp32 WMMA: 16x16x4, K stepped by 4.
// ---------------------------------------------------------------------------
template <int K, bool HAS_AGG, bool RELU>
__global__ __launch_bounds__(256) void gemm_wmma_f32(
    const float* __restrict__ A,
    const float* __restrict__ AGG,
    const float* __restrict__ W,     // [K][64] row-major
    const float* __restrict__ bias,  // [64]
    float* __restrict__ out,         // [n_rows][64]
    int n_rows) {
  __shared__ float sW[K * 64];
  __shared__ float sB[64];

  const int tid = threadIdx.x;
  // Cooperative LDS staging of W (K*64 floats, float4-vectorized).
  for (int i = tid; i < K * 16; i += 256) {
    ((float4*)sW)[i] = ((const float4*)W)[i];
  }
  if (tid < 64) sB[tid] = bias[tid];
  __syncthreads();

  const int wave = tid >> 5;
  const int lane = tid & 31;
  const int row_tile = wave >> 2;            // 0..1
  const int col0 = (wave & 3) * 16;          // 0,16,32,48
  const int row_base = blockIdx.x * 32 + row_tile * 16;

  const int m  = lane & 15;                  // M index within tile (A), N (B/C)
  const int kk = (lane >> 4) * 2;            // lanes 16-31 take K=+2,+3

  // Clamp row for loads (EXEC must be all 1s for WMMA); stores are guarded.
  int row = row_base + m;
  int rowc = (row < n_rows) ? row : 0;
  const float* __restrict__ arow = A + (size_t)rowc * K;
  const float* __restrict__ grow = HAS_AGG ? (AGG + (size_t)rowc * HID) : A;

  v8f c = {};
#pragma unroll
  for (int k = 0; k < K; k += 4) {
    // A-matrix 16x4 fp32 layout: VGPR0 = K=k (lanes0-15) / k+2 (lanes16-31),
    //                            VGPR1 = K=k+1 / k+3  -> contiguous float2.
    v2f a;
    a.x = arow[k + kk + 0];
    a.y = arow[k + kk + 1];
    if (HAS_AGG) {
      a.x += grow[k + kk + 0];
      a.y += grow[k + kk + 1];
    }
    // B-matrix 4x16 fp32, mirrored striping; N = col0 + m from LDS.
    v2f b;
    b.x = sW[(k + kk + 0) * 64 + col0 + m];
    b.y = sW[(k + kk + 1) * 64 + col0 + m];
    c = __builtin_amdgcn_wmma_f32_16x16x4_f32(
        /*neg_a=*/false, a, /*neg_b=*/false, b,
        /*c_mod=*/(short)0, c, /*reuse_a=*/false, /*reuse_b=*/false);
  }

  // C/D layout: VGPR r -> row = r + (lane>>4)*8, col = lane&15.
  const int rbase = row_base + (lane >> 4) * 8;
  const float bv = sB[col0 + m];
#pragma unroll
  for (int r = 0; r < 8; ++r) {
    const int orow = rbase + r;
    if (orow < n_rows) {
      float v = c[r] + bv;
      if (RELU) v = fmaxf(v, 0.0f);
      out[(size_t)orow * HID + col0 + m] = v;
    }
  }
}

// ---------------------------------------------------------------------------
// Edge scatter-add: agg[dst] += h[src].  One wave per edge; lane l handles
// channels l and l+32 -> fully coalesced 128B loads and coalesced atomics.
// ---------------------------------------------------------------------------
__global__ __launch_bounds__(256) void scatter_add_edges(
    const float* __restrict__ h,
    const long long* __restrict__ src,
    const long long* __restrict__ dst,
    float* __restrict__ agg,
    int n_edges) {
  const int e = (int)((blockIdx.x * (unsigned)blockDim.x + threadIdx.x) >> 5);
  const int lane = threadIdx.x & 31;
  if (e >= n_edges) return;
  const long long s = src[e];
  const long long d = dst[e];
  const float* __restrict__ hs = h + (size_t)s * HID;
  float* __restrict__ ad = agg + (size_t)d * HID;
  const float v0 = hs[lane];
  const float v1 = hs[lane + 32];
  unsafeAtomicAdd(&ad[lane], v0);        // global_atomic_add_f32
  unsafeAtomicAdd(&ad[lane + 32], v1);
}

// ---------------------------------------------------------------------------
// Pooling scatter: one wave per node; sums[batch[n]] += h[n]; counts[g] += 1.
// ---------------------------------------------------------------------------
__global__ __launch_bounds__(256) void pool_scatter(
    const float* __restrict__ h,
    const long long* __restrict__ batch,
    float* __restrict__ sums,    // [N_GRAPHS * HID]
    float* __restrict__ counts,  // [N_GRAPHS]
    int n_nodes) {
  const int n = (int)((blockIdx.x * (unsigned)blockDim.x + threadIdx.x) >> 5);
  const int lane = threadIdx.x & 31;
  if (n >= n_nodes) return;
  const long long g = batch[n];
  const float* __restrict__ hr = h + (size_t)n * HID;
  unsafeAtomicAdd(&sums[(size_t)g * HID + lane], hr[lane]);
  unsafeAtomicAdd(&sums[(size_t)g * HID + lane + 32], hr[lane + 32]);
  if (lane == 0) unsafeAtomicAdd(&counts[g], 1.0f);
}

__global__ __launch_bounds__(256) void pool_finalize(
    const float* __restrict__ sums,
    const float* __restrict__ counts,
    float* __restrict__ out) {
  const int i = blockIdx.x * blockDim.x + threadIdx.x;
  if (i < N_GRAPHS * HID) {
    const float cnt = counts[i >> 6];
    out[i] = sums[i] / fmaxf(cnt, 1.0f);
  }
}

// ---------------------------------------------------------------------------
extern "C" void kernel_launch(void* const* d_in, const int* in_sizes, int n_in,
                              void* d_out, int out_size, void* d_ws, size_t ws_size,
                              hipStream_t stream) {
  const float*     x     = (const float*)d_in[0];
  const long long* ei    = (const long long*)d_in[1];
  const long long* batch = (const long long*)d_in[2];
  const float*     Wp    = (const float*)d_in[3];
  const float*     bp    = (const float*)d_in[4];
  const float*     W1_0  = (const float*)d_in[5];
  const float*     b1_0  = (const float*)d_in[6];
  const float*     W2_0  = (const float*)d_in[7];
  const float*     b2_0  = (const float*)d_in[8];
  const float*     W1_1  = (const float*)d_in[9];
  const float*     b1_1  = (const float*)d_in[10];
  const float*     W2_1  = (const float*)d_in[11];
  const float*     b2_1  = (const float*)d_in[12];

  const long long* src = ei;            // edge_index[0]
  const long long* dst = ei + N_EDGES;  // edge_index[1]

  // Workspace layout (all fp32, 256B aligned):
  const size_t node_feat = (size_t)N_NODES * HID;  // 3.2M floats
  char* ws = (char*)d_ws;
  float* h    = (float*)ws;                         ws += ((node_feat * 4 + 255) & ~255ull);
  float* agg  = (float*)ws;                         ws += ((node_feat * 4 + 255) & ~255ull);
  float* tmp  = (float*)ws;                         ws += ((node_feat * 4 + 255) & ~255ull);
  float* sums = (float*)ws;                         ws += (((size_t)N_GRAPHS * HID * 4 + 255) & ~255ull);
  float* cnts = (float*)ws;

  const dim3 blk(256);
  const int gemm_blocks  = (N_NODES + 31) / 32;          // 32 rows per block
  const int edge_blocks  = (N_EDGES + 7) / 8;            // 8 edges (waves) per block
  const int node_blocks  = (N_NODES + 7) / 8;            // 8 nodes (waves) per block

  // h = relu(x @ Wp + bp)
  gemm_wmma_f32<IN_DIM, false, true><<<gemm_blocks, blk, 0, stream>>>(
      x, nullptr, Wp, bp, h, N_NODES);

  // ----- GIN layer 0 -----
  hipMemsetAsync(agg, 0, node_feat * 4, stream);
  scatter_add_edges<<<edge_blocks, blk, 0, stream>>>(h, src, dst, agg, N_EDGES);
  gemm_wmma_f32<HID, true, true><<<gemm_blocks, blk, 0, stream>>>(
      h, agg, W1_0, b1_0, tmp, N_NODES);               // tmp = relu((h+agg)@W1+b1)
  gemm_wmma_f32<HID, false, true><<<gemm_blocks, blk, 0, stream>>>(
      tmp, nullptr, W2_0, b2_0, h, N_NODES);           // h = relu(tmp@W2+b2)

  // ----- GIN layer 1 -----
  hipMemsetAsync(agg, 0, node_feat * 4, stream);
  scatter_add_edges<<<edge_blocks, blk, 0, stream>>>(h, src, dst, agg, N_EDGES);
  gemm_wmma_f32<HID, true, true><<<gemm_blocks, blk, 0, stream>>>(
      h, agg, W1_1, b1_1, tmp, N_NODES);
  gemm_wmma_f32<HID, false, true><<<gemm_blocks, blk, 0, stream>>>(
      tmp, nullptr, W2_1, b2_1, h, N_NODES);

  // ----- global mean pool -----
  hipMemsetAsync(sums, 0, (size_t)N_GRAPHS * HID * 4, stream);
  hipMemsetAsync(cnts, 0, (size_t)N_GRAPHS * 4, stream);
  pool_scatter<<<node_blocks, blk, 0, stream>>>(h, batch, sums, cnts, N_NODES);
  pool_finalize<<<(N_GRAPHS * HID + 255) / 256, blk, 0, stream>>>(
      sums, cnts, (float*)d_out);
}